// RegionalAlignment_86105504350466
// MI455X (gfx1250) — compile-verified
//
#include <hip/hip_runtime.h>
#include <cstddef>

#define B_   64
#define T_   1024
#define N_   256
#define DIM_ 1024
#define K_   4
#define APITCH 260   // padded LDS row pitch for attention (kills bank conflicts)

typedef __attribute__((ext_vector_type(2))) float v2f;
typedef __attribute__((ext_vector_type(8))) float v8f;

__device__ __forceinline__ float wave_red_sum(float v) {
#pragma unroll
  for (int off = 16; off > 0; off >>= 1) v += __shfl_xor(v, off, 32);
  return v;
}

// Segment boundaries from labels = floor(3t/1023): exact integer rationals.
__device__ __constant__ const int c_bnd[K_ + 1] = {0, 341, 682, 1023, 1024};

// ---------------- Kernel 0: L2-normalize the 4 phrase queries -> ws ----------------
__global__ void qnorm_kernel(const float* __restrict__ q, float* __restrict__ qn) {
  const int row  = threadIdx.x >> 5;   // 4 waves, one row each
  const int lane = threadIdx.x & 31;
  float ss = 0.f;
  for (int d = lane; d < DIM_; d += 32) { float x = q[row * DIM_ + d]; ss += x * x; }
  ss = wave_red_sum(ss);
  const float inv = 1.f / fmaxf(sqrtf(ss), 1e-12f);
  for (int d = lane; d < DIM_; d += 32) qn[row * DIM_ + d] = q[row * DIM_ + d] * inv;
}

// ---------------- Kernel 1a: partial segment sums over a T-chunk -> ws ----------------
// grid (DIM/512, B, nch); each chunk writes K_ partial sums per d so the big 256 MB
// stream gets nch x more resident waves (in-flight bytes ~ BW*latency ~ 14 MB).
__global__ void __launch_bounds__(256) event_partial_kernel(const float* __restrict__ audio,
                                                            float* __restrict__ part,
                                                            int tlen) {
  const int b  = blockIdx.y;
  const int ch = blockIdx.z;
  const int d2 = (blockIdx.x * 256 + threadIdx.x) * 2;   // float2 per thread
  const int t0 = ch * tlen, t1 = t0 + tlen;
  const float* base = audio + (size_t)b * T_ * DIM_ + d2;

#pragma unroll
  for (int k = 0; k < K_; ++k) {
    const int lo = max(c_bnd[k], t0), hi = min(c_bnd[k + 1], t1);
    float sx = 0.f, sy = 0.f;
#pragma unroll 4
    for (int t = lo; t < hi; ++t) {
      const float2 v = *reinterpret_cast<const float2*>(base + (size_t)t * DIM_);
      sx += v.x; sy += v.y;
    }
    float* o = part + (((size_t)ch * B_ + b) * K_ + k) * DIM_ + d2;
    o[0] = sx; o[1] = sy;
  }
}

// ---------------- Kernel 1b: reduce chunk partials -> event_tokens ----------------
__global__ void __launch_bounds__(256) event_reduce_kernel(const float* __restrict__ part,
                                                           float* __restrict__ out_evt,
                                                           int nch) {
  const int b  = blockIdx.y;
  const int d2 = (blockIdx.x * 256 + threadIdx.x) * 2;
  const float invc[K_] = {1.f / 341.f, 1.f / 341.f, 1.f / 341.f, 1.f};
#pragma unroll
  for (int k = 0; k < K_; ++k) {
    float sx = 0.f, sy = 0.f;
    for (int ch = 0; ch < nch; ++ch) {
      const float* o = part + (((size_t)ch * B_ + b) * K_ + k) * DIM_ + d2;
      sx += o[0]; sy += o[1];
    }
    float* o = out_evt + ((size_t)b * K_ + k) * DIM_ + d2;
    o[0] = sx * invc[k];
    o[1] = sy * invc[k];
  }
}

// ---------------- Kernel 1 (fallback): single-pass segment mean ----------------
__global__ void __launch_bounds__(256) event_kernel(const float* __restrict__ audio,
                                                    float* __restrict__ out_evt) {
  const int b  = blockIdx.y;
  const int d2 = (blockIdx.x * 256 + threadIdx.x) * 2;
  const float* base = audio + (size_t)b * T_ * DIM_ + d2;
#pragma unroll
  for (int k = 0; k < K_; ++k) {
    float sx = 0.f, sy = 0.f;
#pragma unroll 4
    for (int t = c_bnd[k]; t < c_bnd[k + 1]; ++t) {
      const float2 v = *reinterpret_cast<const float2*>(base + (size_t)t * DIM_);
      sx += v.x; sy += v.y;
    }
    const float invc = 1.f / (float)(c_bnd[k + 1] - c_bnd[k]);
    float* o = out_evt + ((size_t)b * K_ + k) * DIM_ + d2;
    o[0] = sx * invc;
    o[1] = sy * invc;
  }
}

// ---------------- Kernel 2: attention pooling of text -> phrase_tokens ----------------
// One block (8 waves) per batch. Phase 1: dots+norms (coalesced, wave per 32 rows).
// Phase 2: softmax over n in LDS. Phase 3: attn(16pad x 256) x text(256 x 1024) via
// V_WMMA_F32_16X16X4_F32, accumulating K=256 in 64 steps per 16-wide d-tile.
__global__ void __launch_bounds__(256) phrase_kernel(const float* __restrict__ text,
                                                     const float* __restrict__ qn,
                                                     float* __restrict__ out_phr) {
  const int b    = blockIdx.x;
  const int tid  = threadIdx.x;
  const int lane = tid & 31;
  const int wave = tid >> 5;

  __shared__ float qnLds[K_ * DIM_];     // 16 KB
  __shared__ float attnP[16 * APITCH];   // zero-padded A matrix for WMMA, ~16.6 KB
  __shared__ float red[256];

  const float* tb = text + (size_t)b * N_ * DIM_;

  for (int i = tid; i < K_ * DIM_; i += 256) qnLds[i] = qn[i];
  for (int i = tid; i < 16 * APITCH; i += 256) attnP[i] = 0.f;
  __syncthreads();

  // ---- Phase 1: raw dots q_norm . text[n] and ||text[n]|| ----
  for (int r = 0; r < 32; ++r) {
    const int n = wave * 32 + r;
    const float* row = tb + (size_t)n * DIM_;
    float s0 = 0.f, s1 = 0.f, s2 = 0.f, s3 = 0.f, sq = 0.f;
#pragma unroll 4
    for (int d = lane; d < DIM_; d += 32) {
      const float x = row[d];
      sq += x * x;
      s0 += qnLds[d] * x;
      s1 += qnLds[DIM_ + d] * x;
      s2 += qnLds[2 * DIM_ + d] * x;
      s3 += qnLds[3 * DIM_ + d] * x;
    }
    s0 = wave_red_sum(s0); s1 = wave_red_sum(s1);
    s2 = wave_red_sum(s2); s3 = wave_red_sum(s3);
    sq = wave_red_sum(sq);
    if (lane == 0) {
      // score = (q_norm . text_norm)/sqrt(dim) = raw_dot / (||text|| * 32)
      const float inv = 1.f / (fmaxf(sqrtf(sq), 1e-12f) * 32.0f);
      attnP[0 * APITCH + n] = s0 * inv;
      attnP[1 * APITCH + n] = s1 * inv;
      attnP[2 * APITCH + n] = s2 * inv;
      attnP[3 * APITCH + n] = s3 * inv;
    }
  }
  __syncthreads();

  // ---- Phase 2: softmax over n (256) for each of the 4 rows ----
  for (int k = 0; k < K_; ++k) {
    const float v = attnP[k * APITCH + tid];
    red[tid] = v; __syncthreads();
    for (int s = 128; s > 0; s >>= 1) {
      if (tid < s) red[tid] = fmaxf(red[tid], red[tid + s]);
      __syncthreads();
    }
    const float m = red[0]; __syncthreads();
    const float e = expf(v - m);
    red[tid] = e; __syncthreads();
    for (int s = 128; s > 0; s >>= 1) {
      if (tid < s) red[tid] += red[tid + s];
      __syncthreads();
    }
    const float sum = red[0]; __syncthreads();
    attnP[k * APITCH + tid] = e / sum;
    __syncthreads();
  }

  // ---- Phase 3: phrase_tokens = attn x text via fp32 WMMA 16x16x4 ----
  const int arow = lane & 15;
  const int koff = (lane >> 4) << 1;  // lanes 0-15 hold K=0,1 ; lanes 16-31 hold K=2,3
  for (int dt = wave; dt < DIM_ / 16; dt += 8) {
    const int d0 = dt * 16;
    v8f c = {};
    for (int c0 = 0; c0 < N_; c0 += 4) {
      v2f a, bb;
      a.x = attnP[arow * APITCH + c0 + koff];
      a.y = attnP[arow * APITCH + c0 + koff + 1];
      const float* tp = tb + (size_t)(c0 + koff) * DIM_ + d0 + arow;  // 16 contiguous cols/half-wave
      bb.x = tp[0];
      bb.y = tp[DIM_];
      c = __builtin_amdgcn_wmma_f32_16x16x4_f32(false, a, false, bb, (short)0, c,
                                                false, false);
    }
    if (lane < 16) {
#pragma unroll
      for (int m = 0; m < K_; ++m)   // D vgpr m holds out[m][lane] for lanes 0-15
        out_phr[((size_t)b * K_ + m) * DIM_ + d0 + lane] = c[m];
    }
  }
}

// ---------------- Kernel 3: cost matrix + 4x4 Sinkhorn + similarity ----------------
__global__ void __launch_bounds__(32) sinkhorn_kernel(const float* __restrict__ evt,
                                                      const float* __restrict__ phr,
                                                      float* __restrict__ out_sim) {
  const int b    = blockIdx.x;
  const int lane = threadIdx.x;
  const float* e = evt + (size_t)b * K_ * DIM_;
  const float* p = phr + (size_t)b * K_ * DIM_;

  float dot[K_][K_], se[K_], sp[K_];
#pragma unroll
  for (int i = 0; i < K_; ++i) {
    se[i] = 0.f; sp[i] = 0.f;
#pragma unroll
    for (int j = 0; j < K_; ++j) dot[i][j] = 0.f;
  }
  for (int d = lane; d < DIM_; d += 32) {
    float ev[K_], pv[K_];
#pragma unroll
    for (int k = 0; k < K_; ++k) { ev[k] = e[k * DIM_ + d]; pv[k] = p[k * DIM_ + d]; }
#pragma unroll
    for (int k = 0; k < K_; ++k) { se[k] += ev[k] * ev[k]; sp[k] += pv[k] * pv[k]; }
#pragma unroll
    for (int i = 0; i < K_; ++i)
#pragma unroll
      for (int j = 0; j < K_; ++j) dot[i][j] += ev[i] * pv[j];
  }
#pragma unroll
  for (int i = 0; i < K_; ++i) {
    se[i] = wave_red_sum(se[i]);
    sp[i] = wave_red_sum(sp[i]);
#pragma unroll
    for (int j = 0; j < K_; ++j) dot[i][j] = wave_red_sum(dot[i][j]);
  }

  float ie[K_], ip[K_];
#pragma unroll
  for (int i = 0; i < K_; ++i) {
    ie[i] = 1.f / fmaxf(sqrtf(se[i]), 1e-12f);
    ip[i] = 1.f / fmaxf(sqrtf(sp[i]), 1e-12f);
  }
  float cost[K_][K_], Km[K_][K_];
#pragma unroll
  for (int i = 0; i < K_; ++i)
#pragma unroll
    for (int j = 0; j < K_; ++j) {
      cost[i][j] = 1.f - dot[i][j] * ie[i] * ip[j];
      Km[i][j]   = expf(-cost[i][j] * 10.f);   // reg = 0.1
    }

  float u[K_], v[K_];
#pragma unroll
  for (int i = 0; i < K_; ++i) { u[i] = 0.25f; v[i] = 0.25f; }
  for (int it = 0; it < 100; ++it) {
    float us = 0.f;
#pragma unroll
    for (int i = 0; i < K_; ++i) {
      float t = 1e-10f;
#pragma unroll
      for (int j = 0; j < K_; ++j) t += Km[i][j] * v[j];
      u[i] = 1.f / t; us += u[i];
    }
    us += 1e-10f;
#pragma unroll
    for (int i = 0; i < K_; ++i) u[i] /= us;
    float vs = 0.f;
#pragma unroll
    for (int j = 0; j < K_; ++j) {
      float t = 1e-10f;
#pragma unroll
      for (int i = 0; i < K_; ++i) t += Km[i][j] * u[i];
      v[j] = 1.f / t; vs += v[j];
    }
    vs += 1e-10f;
#pragma unroll
    for (int j = 0; j < K_; ++j) v[j] /= vs;
  }

  float sim = 0.f;
#pragma unroll
  for (int i = 0; i < K_; ++i)
#pragma unroll
    for (int j = 0; j < K_; ++j) sim += u[i] * Km[i][j] * v[j] * cost[i][j];

  if (lane == 0) out_sim[b] = -sim;
}

extern "C" void kernel_launch(void* const* d_in, const int* in_sizes, int n_in,
                              void* d_out, int out_size, void* d_ws, size_t ws_size,
                              hipStream_t stream) {
  const float* audio = (const float*)d_in[0];   // (64,1024,1024)
  const float* text  = (const float*)d_in[1];   // (64,256,1024)
  const float* q     = (const float*)d_in[2];   // (4,1024)

  float* out      = (float*)d_out;
  float* out_sim  = out;                                  // (64,)
  float* out_evt  = out + B_;                             // (64,4,1024)
  float* out_phr  = out + B_ + (size_t)B_ * K_ * DIM_;    // (64,4,1024)

  float* qn   = (float*)d_ws;                             // (4,1024) scratch
  float* part = qn + K_ * DIM_;                           // chunk partials (optional)

  const size_t qn_bytes    = (size_t)K_ * DIM_ * sizeof(float);
  const size_t chunk_bytes = (size_t)B_ * K_ * DIM_ * sizeof(float);  // 1 MB per chunk

  // Pick T-chunk count from available workspace (deterministic: ws_size is fixed).
  int nch = 0;
  for (int c = 8; c >= 2; c >>= 1) {
    if (ws_size >= qn_bytes + (size_t)c * chunk_bytes) { nch = c; break; }
  }

  qnorm_kernel<<<1, 128, 0, stream>>>(q, qn);

  if (nch > 0) {
    // Split the 256 MB audio stream across nch T-chunks for more in-flight loads.
    event_partial_kernel<<<dim3(DIM_ / 512, B_, nch), 256, 0, stream>>>(audio, part,
                                                                        T_ / nch);
    event_reduce_kernel<<<dim3(DIM_ / 512, B_), 256, 0, stream>>>(part, out_evt, nch);
  } else {
    event_kernel<<<dim3(DIM_ / 512, B_), 256, 0, stream>>>(audio, out_evt);
  }

  phrase_kernel<<<B_, 256, 0, stream>>>(text, qn, out_phr);
  sinkhorn_kernel<<<B_, 32, 0, stream>>>(out_evt, out_phr, out_sim);
}